// GAT_Net_36335423324692
// MI455X (gfx1250) — compile-verified
//
#include <hip/hip_runtime.h>
#include <hip/hip_bf16.h>
#include <math.h>

typedef __attribute__((ext_vector_type(2))) float v2f;
typedef __attribute__((ext_vector_type(8))) float v8f;

#define NEG_SLOPE 0.2f
#define GAT_EPS 1e-16f
#define ORD_NEG_INF 0x007FFFFFu   // f2ord(-inf)
#define KPAD 168                  // 165 rounded up to multiple of 4

__device__ __forceinline__ unsigned f2ord(float f) {
    unsigned u = __float_as_uint(f);
    return (u & 0x80000000u) ? ~u : (u | 0x80000000u);
}
__device__ __forceinline__ float ord2f(unsigned u) {
    return (u & 0x80000000u) ? __uint_as_float(u & 0x7FFFFFFFu)
                             : __uint_as_float(~u);
}
__device__ __forceinline__ float lrelu(float e) {
    return e > 0.0f ? e : NEG_SLOPE * e;
}

// ---------------------------------------------------------------------------
// Pad+transpose W1[165,256] -> W1t[256,KPAD] (zeros for k >= 165).
// Makes the GEMM inner loop branch-free with contiguous b64 B-fragment loads.
// ---------------------------------------------------------------------------
__global__ void padw1_kernel(const float* __restrict__ W1, float* __restrict__ W1t) {
    int i = blockIdx.x * 256 + threadIdx.x;          // i in [0, 256*KPAD)
    if (i >= 256 * KPAD) return;
    int col = i / KPAD, k = i - col * KPAD;
    W1t[i] = (k < 165) ? W1[k * 256 + col] : 0.0f;
}

// ---------------------------------------------------------------------------
// GEMM1: h1[N,256] = x[N,165] @ W1[165,256], f32 WMMA 16x16x4
// block = 512 threads = 16 waves; one 16-row M tile per block, wave w owns
// columns [16w,16w+16). x tile staged in LDS with K padded 165->168 (zeros).
// B fragments read from padded/transposed W1t (branch-free inner loop).
// ---------------------------------------------------------------------------
__global__ __launch_bounds__(512) void gemm1_wmma_kernel(
    const float* __restrict__ x, const float* __restrict__ W1t,
    float* __restrict__ h1, int N) {
    __shared__ float xs[16 * KPAD];
    const int m0 = blockIdx.x * 16;
    const int tid = threadIdx.x;

    for (int i = tid; i < 16 * KPAD; i += 512) {
        int r = i / KPAD, c = i - r * KPAD;
        int row = m0 + r; if (row >= N) row = N - 1;  // clamp (N%16==0 normally)
        xs[i] = (c < 165) ? x[row * 165 + c] : 0.0f;
    }
    __syncthreads();

    const int wave = tid >> 5;
    const int lane = tid & 31;
    const int n0 = wave * 16;
    const int mrow = lane & 15;
    const int khalf = (lane >> 4) << 1;     // 0 or 2
    const float* arow = xs + mrow * KPAD + khalf;
    const float* brow = W1t + (n0 + mrow) * KPAD + khalf;

    v8f acc = {};
#pragma unroll 6
    for (int k = 0; k < KPAD; k += 4) {
        v2f a, b;
        a.x = arow[k];
        a.y = arow[k + 1];
        float2 bb = *(const float2*)(brow + k);   // rows kb, kb+1 of W1 at col
        b.x = bb.x;
        b.y = bb.y;
        acc = __builtin_amdgcn_wmma_f32_16x16x4_f32(
            false, a, false, b, (short)0, acc, false, false);
    }

    const int chalf = lane >> 4;
    const int col = n0 + (lane & 15);
#pragma unroll
    for (int r = 0; r < 8; ++r) {
        int row = m0 + r + 8 * chalf;
        if (row < N) h1[row * 256 + col] = acc[r];
    }
}

// ---------------------------------------------------------------------------
// Per-(node,head) attention coefficients for layer 1.
// ---------------------------------------------------------------------------
__global__ void attn1_kernel(const float* __restrict__ h1,
                             const float* __restrict__ att_src,
                             const float* __restrict__ att_dst,
                             float* __restrict__ as1, float* __restrict__ ad1,
                             int N) {
    int idx = blockIdx.x * 256 + threadIdx.x;
    if (idx >= N * 4) return;
    int n = idx >> 2, hd = idx & 3;
    const float* hp = h1 + n * 256 + hd * 64;
    const float* sp = att_src + hd * 64;
    const float* dp = att_dst + hd * 64;
    float ts = 0.0f, td = 0.0f;
#pragma unroll 4
    for (int c = 0; c < 64; c += 4) {
        float4 v = *(const float4*)(hp + c);
        float4 a = *(const float4*)(sp + c);
        float4 b = *(const float4*)(dp + c);
        ts += v.x * a.x + v.y * a.y + v.z * a.z + v.w * a.w;
        td += v.x * b.x + v.y * b.y + v.z * b.z + v.w * b.w;
    }
    as1[idx] = ts;
    ad1[idx] = td;
}

// ---------------------------------------------------------------------------
// Init accumulators (ws is poisoned; must re-init every call).
// ---------------------------------------------------------------------------
__global__ void init_ws_kernel(float* __restrict__ out1,
                               unsigned* __restrict__ m1u, float* __restrict__ s1,
                               unsigned* __restrict__ m2u, float* __restrict__ s2,
                               float* __restrict__ out2, int N) {
    int i = blockIdx.x * 256 + threadIdx.x;
    if (i < N * 256) out1[i] = 0.0f;
    if (i < N * 4) { m1u[i] = ORD_NEG_INF; s1[i] = 0.0f; }
    if (i < N) {
        m2u[i] = ORD_NEG_INF; s2[i] = 0.0f;
        out2[2 * i] = 0.0f; out2[2 * i + 1] = 0.0f;
    }
}

__device__ __forceinline__ void edge_nodes(int e, int E, const int* src,
                                           const int* dst, int& s, int& d) {
    if (e < E) { s = src[e]; d = dst[e]; }
    else       { s = e - E;  d = e - E;  }   // self loop
}

// ---------------------------------------------------------------------------
// Layer-1 segment softmax: max pass, sum pass, scatter pass.
// ---------------------------------------------------------------------------
__global__ void pass1_max_kernel(const int* __restrict__ src, const int* __restrict__ dst,
                                 const float* __restrict__ as1, const float* __restrict__ ad1,
                                 unsigned* __restrict__ m1u, int E, int Etot) {
    int e = blockIdx.x * 256 + threadIdx.x;
    if (e >= Etot) return;
    int s, d; edge_nodes(e, E, src, dst, s, d);
    float4 a = *(const float4*)(as1 + 4 * s);
    float4 b = *(const float4*)(ad1 + 4 * d);
    atomicMax(m1u + 4 * d + 0, f2ord(lrelu(a.x + b.x)));
    atomicMax(m1u + 4 * d + 1, f2ord(lrelu(a.y + b.y)));
    atomicMax(m1u + 4 * d + 2, f2ord(lrelu(a.z + b.z)));
    atomicMax(m1u + 4 * d + 3, f2ord(lrelu(a.w + b.w)));
}

__global__ void pass1_sum_kernel(const int* __restrict__ src, const int* __restrict__ dst,
                                 const float* __restrict__ as1, const float* __restrict__ ad1,
                                 const unsigned* __restrict__ m1u, float* __restrict__ s1,
                                 int E, int Etot) {
    int e = blockIdx.x * 256 + threadIdx.x;
    if (e >= Etot) return;
    int s, d; edge_nodes(e, E, src, dst, s, d);
    float4 a = *(const float4*)(as1 + 4 * s);
    float4 b = *(const float4*)(ad1 + 4 * d);
    float ev[4] = {a.x + b.x, a.y + b.y, a.z + b.z, a.w + b.w};
#pragma unroll
    for (int h = 0; h < 4; ++h) {
        float t = lrelu(ev[h]);
        float m = ord2f(m1u[4 * d + h]);
        atomicAdd(s1 + 4 * d + h, __expf(t - m));
    }
}

// wave (32 lanes) per edge: 256 channels, 8 per lane
__global__ void pass1_scatter_kernel(const int* __restrict__ src, const int* __restrict__ dst,
                                     const float* __restrict__ as1, const float* __restrict__ ad1,
                                     const unsigned* __restrict__ m1u, const float* __restrict__ s1,
                                     const float* __restrict__ h1, float* __restrict__ out1,
                                     int E, int Etot) {
    int e = (blockIdx.x * 256 + threadIdx.x) >> 5;
    if (e >= Etot) return;
    int lane = threadIdx.x & 31;
    int s, d; edge_nodes(e, E, src, dst, s, d);
    float alpha[4];
#pragma unroll
    for (int h = 0; h < 4; ++h) {
        float t = lrelu(as1[4 * s + h] + ad1[4 * d + h]);
        float m = ord2f(m1u[4 * d + h]);
        alpha[h] = __expf(t - m) / (s1[4 * d + h] + GAT_EPS);
    }
    const float* hs = h1 + s * 256;
    float* od = out1 + d * 256;
#pragma unroll
    for (int i = 0; i < 8; ++i) {
        int c = lane + 32 * i;
        atomicAdd(od + c, hs[c] * alpha[c >> 6]);
    }
}

// ---------------------------------------------------------------------------
// ReLU(out1 + b1) in place; fused tiny GEMM2 (256->2) + layer-2 attn coefs.
// One wave per node.
// ---------------------------------------------------------------------------
__global__ void finish1_kernel(float* __restrict__ out1, const float* __restrict__ b1,
                               const float* __restrict__ W2,
                               const float* __restrict__ att_src2,
                               const float* __restrict__ att_dst2,
                               float* __restrict__ h2s, float* __restrict__ as2,
                               float* __restrict__ ad2, int N) {
    int n = (blockIdx.x * 256 + threadIdx.x) >> 5;
    if (n >= N) return;
    int lane = threadIdx.x & 31;
    float* row = out1 + n * 256;
    float t0 = 0.0f, t1 = 0.0f;
#pragma unroll
    for (int i = 0; i < 8; ++i) {
        int c = lane + 32 * i;
        float v = fmaxf(row[c] + b1[c], 0.0f);
        row[c] = v;
        t0 += v * W2[2 * c];
        t1 += v * W2[2 * c + 1];
    }
#pragma unroll
    for (int off = 16; off > 0; off >>= 1) {
        t0 += __shfl_xor(t0, off, 32);
        t1 += __shfl_xor(t1, off, 32);
    }
    if (lane == 0) {
        h2s[2 * n] = t0; h2s[2 * n + 1] = t1;
        as2[n] = t0 * att_src2[0] + t1 * att_src2[1];
        ad2[n] = t0 * att_dst2[0] + t1 * att_dst2[1];
    }
}

// ---------------------------------------------------------------------------
// Layer-2 (1 head, 2 channels) softmax passes: thread per edge.
// ---------------------------------------------------------------------------
__global__ void pass2_max_kernel(const int* __restrict__ src, const int* __restrict__ dst,
                                 const float* __restrict__ as2, const float* __restrict__ ad2,
                                 unsigned* __restrict__ m2u, int E, int Etot) {
    int e = blockIdx.x * 256 + threadIdx.x;
    if (e >= Etot) return;
    int s, d; edge_nodes(e, E, src, dst, s, d);
    atomicMax(m2u + d, f2ord(lrelu(as2[s] + ad2[d])));
}

__global__ void pass2_sum_kernel(const int* __restrict__ src, const int* __restrict__ dst,
                                 const float* __restrict__ as2, const float* __restrict__ ad2,
                                 const unsigned* __restrict__ m2u, float* __restrict__ s2,
                                 int E, int Etot) {
    int e = blockIdx.x * 256 + threadIdx.x;
    if (e >= Etot) return;
    int s, d; edge_nodes(e, E, src, dst, s, d);
    float t = lrelu(as2[s] + ad2[d]);
    atomicAdd(s2 + d, __expf(t - ord2f(m2u[d])));
}

__global__ void pass2_scatter_kernel(const int* __restrict__ src, const int* __restrict__ dst,
                                     const float* __restrict__ as2, const float* __restrict__ ad2,
                                     const unsigned* __restrict__ m2u, const float* __restrict__ s2,
                                     const float* __restrict__ h2s, float* __restrict__ out2,
                                     int E, int Etot) {
    int e = blockIdx.x * 256 + threadIdx.x;
    if (e >= Etot) return;
    int s, d; edge_nodes(e, E, src, dst, s, d);
    float t = lrelu(as2[s] + ad2[d]);
    float alpha = __expf(t - ord2f(m2u[d])) / (s2[d] + GAT_EPS);
    atomicAdd(out2 + 2 * d,     h2s[2 * s]     * alpha);
    atomicAdd(out2 + 2 * d + 1, h2s[2 * s + 1] * alpha);
}

__global__ void finalize_kernel(const float* __restrict__ out2, const float* __restrict__ b2,
                                float* __restrict__ out, int N) {
    int n = blockIdx.x * 256 + threadIdx.x;
    if (n >= N) return;
    float o0 = out2[2 * n] + b2[0];
    float o1 = out2[2 * n + 1] + b2[1];
    float mx = fmaxf(o0, o1);
    float l = logf(expf(o0 - mx) + expf(o1 - mx));
    out[2 * n] = o0 - mx - l;
    out[2 * n + 1] = o1 - mx - l;
}

// ---------------------------------------------------------------------------
extern "C" void kernel_launch(void* const* d_in, const int* in_sizes, int n_in,
                              void* d_out, int out_size, void* d_ws, size_t ws_size,
                              hipStream_t stream) {
    const float* x        = (const float*)d_in[0];
    const int*   src      = (const int*)  d_in[1];
    const int*   dst      = (const int*)  d_in[2];
    const float* W1       = (const float*)d_in[3];
    const float* att_src1 = (const float*)d_in[4];
    const float* att_dst1 = (const float*)d_in[5];
    const float* b1       = (const float*)d_in[6];
    const float* W2       = (const float*)d_in[7];
    const float* att_src2 = (const float*)d_in[8];
    const float* att_dst2 = (const float*)d_in[9];
    const float* b2       = (const float*)d_in[10];
    float* out = (float*)d_out;

    const int N = in_sizes[0] / 165;
    const int E = in_sizes[1];
    const int Etot = E + N;            // with self loops

    // workspace layout (floats)
    float* ws  = (float*)d_ws;
    float* h1   = ws;                         // N*256
    float* out1 = h1 + (size_t)N * 256;       // N*256 (becomes h2 in place)
    float* as1  = out1 + (size_t)N * 256;     // N*4
    float* ad1  = as1 + (size_t)N * 4;        // N*4
    unsigned* m1u = (unsigned*)(ad1 + (size_t)N * 4);  // N*4
    float* s1   = (float*)(m1u + (size_t)N * 4);       // N*4
    float* h2s  = s1 + (size_t)N * 4;         // N*2
    float* as2  = h2s + (size_t)N * 2;        // N
    float* ad2  = as2 + N;                    // N
    unsigned* m2u = (unsigned*)(ad2 + N);     // N
    float* s2   = (float*)(m2u + N);          // N
    float* out2 = s2 + N;                     // N*2
    float* W1t  = out2 + (size_t)N * 2;       // 256*KPAD padded/transposed W1

    const int mtiles   = (N + 15) / 16;
    const int eblocks  = (Etot + 255) / 256;
    const int ewblocks = (Etot + 7) / 8;      // wave per edge, 8 waves/block
    const int nwblocks = (N + 7) / 8;         // wave per node

    padw1_kernel<<<(256 * KPAD + 255) / 256, 256, 0, stream>>>(W1, W1t);
    gemm1_wmma_kernel<<<mtiles, 512, 0, stream>>>(x, W1t, h1, N);
    attn1_kernel<<<(N * 4 + 255) / 256, 256, 0, stream>>>(h1, att_src1, att_dst1, as1, ad1, N);
    init_ws_kernel<<<N, 256, 0, stream>>>(out1, m1u, s1, m2u, s2, out2, N);

    pass1_max_kernel<<<eblocks, 256, 0, stream>>>(src, dst, as1, ad1, m1u, E, Etot);
    pass1_sum_kernel<<<eblocks, 256, 0, stream>>>(src, dst, as1, ad1, m1u, s1, E, Etot);
    pass1_scatter_kernel<<<ewblocks, 256, 0, stream>>>(src, dst, as1, ad1, m1u, s1, h1, out1, E, Etot);

    finish1_kernel<<<nwblocks, 256, 0, stream>>>(out1, b1, W2, att_src2, att_dst2, h2s, as2, ad2, N);

    pass2_max_kernel<<<eblocks, 256, 0, stream>>>(src, dst, as2, ad2, m2u, E, Etot);
    pass2_sum_kernel<<<eblocks, 256, 0, stream>>>(src, dst, as2, ad2, m2u, s2, E, Etot);
    pass2_scatter_kernel<<<eblocks, 256, 0, stream>>>(src, dst, as2, ad2, m2u, s2, h2s, out2, E, Etot);

    finalize_kernel<<<(N + 255) / 256, 256, 0, stream>>>(out2, b2, out, N);
}